// Unwrapper_36326833389839
// MI455X (gfx1250) — compile-verified
//
#include <hip/hip_runtime.h>
#include <stdint.h>

#define NVERT 499998
#define NFACE 1000000

typedef unsigned int u32x4 __attribute__((ext_vector_type(4)));
typedef int          i32x4 __attribute__((ext_vector_type(4)));
typedef int          i32x8 __attribute__((ext_vector_type(8)));
typedef float        v2f   __attribute__((ext_vector_type(2)));
typedef float        v8f   __attribute__((ext_vector_type(8)));

// ---------------------------------------------------------------------------
// TDM: DMA a contiguous run of int32 (<=768 elems) from global into LDS.
// D# group0: [1:0]=count=1, [63:32]=lds_addr, [120:64]=global_addr, [127:126]=type=2
// D# group1: [17:16]=data_size(2 -> 4B), [79:48]=tensor_dim0 (=valid elems, OOB reads 0),
//            [111:80]=tensor_dim1(=1), [127:112]=tile_dim0(=768), [207:160]=dim0_stride
// ---------------------------------------------------------------------------
__device__ __forceinline__ void tdm_stage_tri(const int* gsrc, int valid_elems,
                                              uint32_t lds_byte_addr) {
  uint64_t ga = (uint64_t)(uintptr_t)gsrc;
  u32x4 g0;
  g0.x = 1u;                                              // count=1, user mode
  g0.y = lds_byte_addr;                                   // LDS dest (bytes)
  g0.z = (uint32_t)ga;                                    // global_addr[31:0]
  g0.w = (uint32_t)((ga >> 32) & 0x01FFFFFFull) | 0x80000000u; // addr[56:32] | type=2
  uint32_t td0 = (uint32_t)valid_elems;
  i32x8 g1;
  g1[0] = (int)(2u << 16);                 // workgroup_mask=0, data_size=4B
  g1[1] = (int)((td0 & 0xFFFFu) << 16);    // tensor_dim0[15:0]
  g1[2] = (int)((td0 >> 16) | (1u << 16)); // tensor_dim0[31:16], tensor_dim1=1
  g1[3] = (int)(768u << 16);               // tile_dim0 = 768
  g1[4] = 0;                               // tile_dim1/2 unused
  g1[5] = 768;                             // tensor_dim0_stride
  g1[6] = 0;
  g1[7] = 0;
  i32x4 z4 = (i32x4)0;
#if defined(__clang_major__) && (__clang_major__ >= 23)
  i32x8 z8 = (i32x8)0;
  __builtin_amdgcn_tensor_load_to_lds(g0, g1, z4, z4, z8, 0);
#else
  __builtin_amdgcn_tensor_load_to_lds(g0, g1, z4, z4, 0);
#endif
}

// ---------------------------------------------------------------------------
// face_normal @ axes.T via V_WMMA_F32_16X16X4_F32, transposed so N = face:
//   D(16x16) = A(axes, 16x4: M=axis, K=xyz0) x B(normals^T, 4x16: K=xyz0, N=face)
// A layout (32-bit A 16x4): lanes 0-15 hold row M=lane, VGPR0=K0, VGPR1=K1;
//                           lanes 16-31 hold K2/K3 for M=lane-16.  (constants)
// B layout (mirrors A):     VGPR0: lanes 0-15 = row K0 (N=lane), lanes 16-31 = K2;
//                           VGPR1: K1 / K3.   (normals broadcast via shuffles)
// D layout: VGPR r, lane n(0-15) = (M=r, N=n)  ->  lane n holds face n's 6 dots
// in D[0..5] of its OWN lane; argmax is a per-lane 5-compare loop (first-max
// tie-break like jnp.argmax; zero-padded columns lose ties since max >= 0).
// EXEC must be all 1s: callers run this unconditionally (no early returns).
// ---------------------------------------------------------------------------
__device__ __forceinline__ int wmma_axis_index(float nx, float ny, float nz) {
  const int lane = (int)(threadIdx.x & 31u);
  const int nsub = lane & 15;
  // A = axes matrix rows: 0:(1,0,0) 1:(-1,0,0) 2:(0,1,0) 3:(0,-1,0) 4:(0,0,1) 5:(0,0,-1)
  float a0, a1;
  if (lane < 16) {  // K=0 (x comp), K=1 (y comp) of axis row M=nsub
    a0 = (nsub == 0) ? 1.0f : (nsub == 1) ? -1.0f : 0.0f;
    a1 = (nsub == 2) ? 1.0f : (nsub == 3) ? -1.0f : 0.0f;
  } else {          // K=2 (z comp), K=3 (pad)
    a0 = (nsub == 4) ? 1.0f : (nsub == 5) ? -1.0f : 0.0f;
    a1 = 0.0f;
  }
  v2f A; A.x = a0; A.y = a1;

  int index = 0;
#pragma unroll
  for (int g = 0; g < 2; ++g) {          // 32 faces/wave -> two 16-face WMMAs
    const int src = g * 16 + nsub;       // lane owning face g*16+nsub
    float bx = __shfl(nx, src, 32);      // unconditional shuffles (full EXEC)
    float by = __shfl(ny, src, 32);
    float bz = __shfl(nz, src, 32);
    v2f B;
    B.x = (lane < 16) ? bx : bz;         // K0 row : K2 row
    B.y = (lane < 16) ? by : 0.0f;       // K1 row : K3 pad
    v8f C = {0.f, 0.f, 0.f, 0.f, 0.f, 0.f, 0.f, 0.f};
    // F32 WMMA: NEG must be {CNeg,0,0} -> neg_a/neg_b = false
    v8f D = __builtin_amdgcn_wmma_f32_16x16x4_f32(false, A, false, B,
                                                  (short)0, C, false, false);
    int bi = 0;
    float bv = D[0];
#pragma unroll
    for (int j = 1; j < 6; ++j) {        // first-max tie-break (strict >)
      float dj = D[j];
      if (dj > bv) { bv = dj; bi = j; }
    }
    int got = __shfl(bi, nsub, 32);      // fetch result for our face from lane n
    if ((lane >> 4) == g) index = got;
  }
  return index;
}

// ---------------------------------------------------------------------------
// Kernel 0: zero accumulators (tangent region of d_out, per-vertex counts, gmax)
// ---------------------------------------------------------------------------
__global__ void __launch_bounds__(256) k_init(float* __restrict__ tang_out,
                                              float* __restrict__ counts,
                                              unsigned* __restrict__ gmax) {
  int i = blockIdx.x * 256 + threadIdx.x;
  if (i < NVERT * 3) tang_out[i] = 0.0f;
  if (i < NVERT) counts[i] = 0.0f;
  if (i < 3) gmax[i] = 0u;   // values are >=0 so float bits order as uint
}

// ---------------------------------------------------------------------------
// Kernel 1: per-face max_axis -> LDS max -> global atomicMax (3 scalars)
// ---------------------------------------------------------------------------
__global__ void __launch_bounds__(256) k_pass1(const float* __restrict__ vp,
                                               const float* __restrict__ vnrm,
                                               const int* __restrict__ tri,
                                               const float* __restrict__ bbox,
                                               unsigned* __restrict__ gmax) {
  __shared__ int stri[768];
  __shared__ unsigned smax[3];
  const int f0 = blockIdx.x * 256;
  if (threadIdx.x == 0) {
    int base = f0 * 3;
    int valid = 3 * NFACE - base;
    if (valid > 768) valid = 768;
    tdm_stage_tri(tri + base, valid, (uint32_t)(uintptr_t)&stri[0]);
    __builtin_amdgcn_s_wait_tensorcnt(0);
    smax[0] = 0u; smax[1] = 0u; smax[2] = 0u;
  }
  __syncthreads();

  const int f = f0 + (int)threadIdx.x;
  const bool active = (f < NFACE);

  int id[3];
#pragma unroll
  for (int c = 0; c < 3; ++c) id[c] = stri[threadIdx.x * 3 + c]; // tail: TDM OOB zeros

  const float bx0 = bbox[0], by0 = bbox[1], bz0 = bbox[2];
  const float sx2 = 2.0f / (bbox[3] - bx0);
  const float sy2 = 2.0f / (bbox[4] - by0);
  const float sz2 = 2.0f / (bbox[5] - bz0);

  float vx[3], vy[3], vz[3];
  float nx = 0.f, ny = 0.f, nz = 0.f;
#pragma unroll
  for (int c = 0; c < 3; ++c) {
    int iv = id[c];
    vx[c] = (vp[iv * 3 + 0] - bx0) * sx2 - 1.0f;
    vy[c] = (vp[iv * 3 + 1] - by0) * sy2 - 1.0f;
    vz[c] = (vp[iv * 3 + 2] - bz0) * sz2 - 1.0f;
    nx += vnrm[iv * 3 + 0];
    ny += vnrm[iv * 3 + 1];
    nz += vnrm[iv * 3 + 2];
  }
  // normalize(.) is a positive scale -> argmax unchanged; WMMA needs full EXEC
  int index = wmma_axis_index(nx, ny, nz);

  float ma[3] = {0.0f, 0.0f, 0.0f};
  if (active) {
#pragma unroll
    for (int c = 0; c < 3; ++c)
      ma[c] = (index <= 1) ? fabsf(vx[c]) : (index <= 3) ? fabsf(vy[c]) : fabsf(vz[c]);
  }
#pragma unroll
  for (int c = 0; c < 3; ++c) atomicMax(&smax[c], __float_as_uint(ma[c]));
  __syncthreads();
  if (threadIdx.x < 3) atomicMax(&gmax[threadIdx.x], smax[threadIdx.x]);
}

// ---------------------------------------------------------------------------
// Kernel 2: finalize uv/index, compute tangent, scatter-add to vertices
// ---------------------------------------------------------------------------
__global__ void __launch_bounds__(256) k_pass2(const float* __restrict__ vp,
                                               const float* __restrict__ vnrm,
                                               const int* __restrict__ tri,
                                               const float* __restrict__ bbox,
                                               const float* __restrict__ gmax,
                                               float* __restrict__ uv_out,
                                               float* __restrict__ idx_out,
                                               float* __restrict__ tang_out,
                                               float* __restrict__ counts) {
  __shared__ int stri[768];
  const int f0 = blockIdx.x * 256;
  if (threadIdx.x == 0) {
    int base = f0 * 3;
    int valid = 3 * NFACE - base;
    if (valid > 768) valid = 768;
    tdm_stage_tri(tri + base, valid, (uint32_t)(uintptr_t)&stri[0]);
    __builtin_amdgcn_s_wait_tensorcnt(0);
  }
  __syncthreads();

  const int f = f0 + (int)threadIdx.x;
  const bool active = (f < NFACE);   // no early return: WMMA needs full EXEC

  int id[3];
#pragma unroll
  for (int c = 0; c < 3; ++c) id[c] = stri[threadIdx.x * 3 + c]; // tail: zeros

  const float bx0 = bbox[0], by0 = bbox[1], bz0 = bbox[2];
  const float sx2 = 2.0f / (bbox[3] - bx0);
  const float sy2 = 2.0f / (bbox[4] - by0);
  const float sz2 = 2.0f / (bbox[5] - bz0);

  float vx[3], vy[3], vz[3];
  float px[3], py[3], pz[3];
  float nx = 0.f, ny = 0.f, nz = 0.f;
#pragma unroll
  for (int c = 0; c < 3; ++c) {
    int iv = id[c];
    px[c] = vp[iv * 3 + 0];
    py[c] = vp[iv * 3 + 1];
    pz[c] = vp[iv * 3 + 2];
    vx[c] = (px[c] - bx0) * sx2 - 1.0f;
    vy[c] = (py[c] - by0) * sy2 - 1.0f;
    vz[c] = (pz[c] - bz0) * sz2 - 1.0f;
    nx += vnrm[iv * 3 + 0];
    ny += vnrm[iv * 3 + 1];
    nz += vnrm[iv * 3 + 2];
  }
  int index = wmma_axis_index(nx, ny, nz);

  float uc[3], vc[3];
#pragma unroll
  for (int c = 0; c < 3; ++c) {
    float ur = (index <= 1) ? vy[c] : vx[c];
    float vr = (index <= 3) ? -vz[c] : ((index == 4) ? vy[c] : -vy[c]);
    float ginv = 1.0f / gmax[c];   // per-CORNER global max (axis=0 reduction)
    float u = fminf(fmaxf((ur * ginv + 1.0f) * 0.5f, 0.0f), 1.0f);
    float w = fminf(fmaxf((vr * ginv + 1.0f) * 0.5f, 0.0f), 1.0f);
    uc[c] = u; vc[c] = w;
    if (active) {
      uv_out[(size_t)f * 6 + c * 2 + 0] = u;
      uv_out[(size_t)f * 6 + c * 2 + 1] = w;
    }
  }
  if (active) idx_out[f] = (float)index;

  float du1x = uc[1] - uc[0], du1y = vc[1] - vc[0];
  float du2x = uc[2] - uc[0], du2y = vc[2] - vc[0];
  float d1x = px[1] - px[0], d1y = py[1] - py[0], d1z = pz[1] - pz[0];
  float d2x = px[2] - px[0], d2y = py[2] - py[0], d2z = pz[2] - pz[0];
  float denom = fmaxf(du1x * du2y - du1y * du2x, 1e-6f);  // jnp.clip(denom, 1e-6)
  float inv = 1.0f / denom;
  float tx = (d1x * du2y - d2x * du1y) * inv;
  float ty = (d1y * du2y - d2y * du1y) * inv;
  float tz = (d1z * du2y - d2z * du1y) * inv;

  if (active) {
#pragma unroll
    for (int c = 0; c < 3; ++c) {
      int iv = id[c];
      atomicAdd(&tang_out[iv * 3 + 0], tx);
      atomicAdd(&tang_out[iv * 3 + 1], ty);
      atomicAdd(&tang_out[iv * 3 + 2], tz);
      atomicAdd(&counts[iv], 1.0f);   // tansum has equal components == corner count
    }
  }
}

// ---------------------------------------------------------------------------
// Kernel 3: per-vertex normalize + Gram-Schmidt against vertex normal
// ---------------------------------------------------------------------------
__global__ void __launch_bounds__(256) k_pass3(const float* __restrict__ vnrm,
                                               const float* __restrict__ counts,
                                               float* __restrict__ tang) {
  int v = blockIdx.x * 256 + threadIdx.x;
  if (v >= NVERT) return;
  float c = counts[v];
  float tx = 0.f, ty = 0.f, tz = 0.f;
  if (c > 0.0f) {
    float cinv = 1.0f / c;
    tx = tang[v * 3 + 0] * cinv;
    ty = tang[v * 3 + 1] * cinv;
    tz = tang[v * 3 + 2] * cinv;
  }
  float n1 = sqrtf(tx * tx + ty * ty + tz * tz);
  float i1 = 1.0f / fmaxf(n1, 1e-12f);
  tx *= i1; ty *= i1; tz *= i1;

  float nx = vnrm[v * 3 + 0], ny = vnrm[v * 3 + 1], nz = vnrm[v * 3 + 2];
  float d = tx * nx + ty * ny + tz * nz;
  tx -= d * nx; ty -= d * ny; tz -= d * nz;
  float n2 = sqrtf(tx * tx + ty * ty + tz * tz);
  float i2 = 1.0f / fmaxf(n2, 1e-12f);
  tang[v * 3 + 0] = tx * i2;
  tang[v * 3 + 1] = ty * i2;
  tang[v * 3 + 2] = tz * i2;
}

// ---------------------------------------------------------------------------
extern "C" void kernel_launch(void* const* d_in, const int* in_sizes, int n_in,
                              void* d_out, int out_size, void* d_ws, size_t ws_size,
                              hipStream_t stream) {
  (void)in_sizes; (void)n_in; (void)out_size; (void)ws_size;
  const float* vp   = (const float*)d_in[0];  // (NV,3) f32
  const float* vn   = (const float*)d_in[1];  // (NV,3) f32
  const int*   tri  = (const int*)d_in[2];    // (NF,3) i32
  const float* bbox = (const float*)d_in[3];  // (2,3) f32

  float* out      = (float*)d_out;
  float* uv_out   = out;                       // NF*3*2
  float* idx_out  = out + (size_t)NFACE * 6;   // NF (index as float)
  float* tang_out = idx_out + NFACE;           // NV*3

  unsigned* gmax_u = (unsigned*)d_ws;          // 3 words (float bits)
  float* counts    = (float*)d_ws + 16;        // NV floats

  const int nface_blk = (NFACE + 255) / 256;
  k_init<<<(NVERT * 3 + 255) / 256, 256, 0, stream>>>(tang_out, counts, gmax_u);
  k_pass1<<<nface_blk, 256, 0, stream>>>(vp, vn, tri, bbox, gmax_u);
  k_pass2<<<nface_blk, 256, 0, stream>>>(vp, vn, tri, bbox, (const float*)d_ws,
                                         uv_out, idx_out, tang_out, counts);
  k_pass3<<<(NVERT + 255) / 256, 256, 0, stream>>>(vn, counts, tang_out);
}